// LESPLoss_79834852098555
// MI455X (gfx1250) — compile-verified
//
#include <hip/hip_runtime.h>
#include <math.h>

// LESP loss: per row, loss = softplus(logsumexp(neg logits) + logsumexp(-pos logits)),
// mean over rows having both a positive and a negative. Memory-bound streaming reduction:
// 256 MiB @ 23.3 TB/s => ~11.5us floor; keep exactly 1 v_exp_f32 per element.

#define NEG_INF_F (-1e30f)
#define LCOLS     4096
#define TPB       256
#define NWAVES    (TPB / 32)          // 8 waves (wave32)
#define PER_WAVE  (LCOLS / NWAVES)    // 512 elements per wave
#define PER_LANE  (PER_WAVE / 32)     // 16 elements per lane

// gfx12-style cache policy: TH in cpol[2:0]; TH_NT = 1 (single-pass stream,
// 256 MiB working set > 192 MB L2 -> don't fight for L2 residency).
#define CPOL_NT   1

typedef __attribute__((ext_vector_type(4))) float f32x4;
typedef __attribute__((ext_vector_type(4))) int   i32x4;

// CDNA5 async global->LDS path (ASYNCcnt). Guarded: falls back to direct NT loads
// if this toolchain does not expose the builtin.
#if defined(__HIP_DEVICE_COMPILE__) && __has_builtin(__builtin_amdgcn_global_load_async_to_lds_b128)
#define USE_ASYNC_LDS 1
#else
#define USE_ASYNC_LDS 0
#endif

__device__ __forceinline__ void lse_combine(float& m, float& s, float m2, float s2) {
    // merge two (max, sum-exp-rel-max) partials; exact when either side is empty (m=-1e30, s=0)
    float M = fmaxf(m, m2);
    s = s * __expf(m - M) + s2 * __expf(m2 - M);
    m = M;
}

__global__ void __launch_bounds__(TPB) lesp_row_kernel(
    const float* __restrict__ logits,
    const int*   __restrict__ targets,
    float* __restrict__ row_loss,
    float* __restrict__ row_valid)
{
    const int row  = blockIdx.x;
    const int tid  = threadIdx.x;
    const int lane = tid & 31;
    const int wave = tid >> 5;

    const float* lrow = logits  + (size_t)row * LCOLS;
    const int*   trow = targets + (size_t)row * LCOLS;

    float    xs[PER_LANE];
    unsigned selmask = 0;   // bit j set => element j is a positive (use -x stream)

#if USE_ASYNC_LDS
    __shared__ float s_log[LCOLS];
    __shared__ int   s_tgt[LCOLS];
    {
        // Builtin wants pointers to 16-byte vectors: global src (as1), LDS dst (as3).
        typedef __attribute__((address_space(1))) i32x4 gvec_t;
        typedef __attribute__((address_space(3))) i32x4 lvec_t;
        gvec_t* gl = (gvec_t*)lrow;
        gvec_t* gt = (gvec_t*)trow;
        lvec_t* ll = (lvec_t*)s_log;
        lvec_t* lt = (lvec_t*)s_tgt;
        // Each wave stages its own 2KB slice per stream: 4 async b128 ops
        // (32 lanes x 16B contiguous per op), addresses in 16B units.
        const int v0 = wave * (PER_WAVE / 4) + lane;   // float4 index
        #pragma unroll
        for (int i = 0; i < 4; ++i) {
            const int v = v0 + i * 32;
            __builtin_amdgcn_global_load_async_to_lds_b128(gl + v, ll + v, 0, CPOL_NT);
            __builtin_amdgcn_global_load_async_to_lds_b128(gt + v, lt + v, 0, CPOL_NT);
        }
    }
#if __has_builtin(__builtin_amdgcn_s_wait_asynccnt)
    __builtin_amdgcn_s_wait_asynccnt(0);
#else
    asm volatile("s_wait_asynccnt 0" ::: "memory");
#endif
    asm volatile("" ::: "memory");
    // Each wave reads back only the slice it staged itself -> no workgroup barrier.
    // Stride-32 b32 DS reads: bank-conflict free.
    #pragma unroll
    for (int j = 0; j < PER_LANE; ++j) {
        const int col = wave * PER_WAVE + j * 32 + lane;
        xs[j] = s_log[col];
        selmask |= (s_tgt[col] > 0) ? (1u << j) : 0u;
    }
#else
    // Direct path: coalesced nontemporal b128 loads (streamed once; skip L2 retention).
    const f32x4* lrow4 = (const f32x4*)lrow;
    const i32x4* trow4 = (const i32x4*)trow;
    #pragma unroll
    for (int q = 0; q < 4; ++q) {
        const int idx = wave * (PER_WAVE / 4) + q * 32 + lane;
        const f32x4 x4 = __builtin_nontemporal_load(&lrow4[idx]);
        const i32x4 t4 = __builtin_nontemporal_load(&trow4[idx]);
        xs[q * 4 + 0] = x4.x;  selmask |= (t4.x > 0) ? (1u << (q * 4 + 0)) : 0u;
        xs[q * 4 + 1] = x4.y;  selmask |= (t4.y > 0) ? (1u << (q * 4 + 1)) : 0u;
        xs[q * 4 + 2] = x4.z;  selmask |= (t4.z > 0) ? (1u << (q * 4 + 2)) : 0u;
        xs[q * 4 + 3] = x4.w;  selmask |= (t4.w > 0) ? (1u << (q * 4 + 3)) : 0u;
    }
#endif

    // Pass 1: per-thread stream maxes (no transcendentals, pure v_max/v_cndmask).
    float m_pos = NEG_INF_F, m_neg = NEG_INF_F;
    #pragma unroll
    for (int j = 0; j < PER_LANE; ++j) {
        const bool  p = (selmask >> j) & 1u;
        const float v = p ? -xs[j] : xs[j];
        m_pos = p ? fmaxf(m_pos, v) : m_pos;
        m_neg = p ? m_neg : fmaxf(m_neg, v);
    }

    // Pass 2: exactly one v_exp_f32 per element, branchless accumulator select.
    float s_pos = 0.0f, s_neg = 0.0f;
    #pragma unroll
    for (int j = 0; j < PER_LANE; ++j) {
        const bool  p = (selmask >> j) & 1u;
        const float v = p ? -xs[j] : xs[j];
        const float m = p ? m_pos : m_neg;
        const float e = __expf(v - m);   // exponent <= 0, never overflows
        s_pos += p ? e : 0.0f;
        s_neg += p ? 0.0f : e;
    }

    // Wave32 tree combine (5 shuffle steps).
    #pragma unroll
    for (int off = 16; off > 0; off >>= 1) {
        float m2 = __shfl_xor(m_neg, off, 32);
        float s2 = __shfl_xor(s_neg, off, 32);
        lse_combine(m_neg, s_neg, m2, s2);
        m2 = __shfl_xor(m_pos, off, 32);
        s2 = __shfl_xor(s_pos, off, 32);
        lse_combine(m_pos, s_pos, m2, s2);
    }

    // Cross-wave combine through LDS.
    __shared__ float red[NWAVES][4];
    if (lane == 0) {
        red[wave][0] = m_neg; red[wave][1] = s_neg;
        red[wave][2] = m_pos; red[wave][3] = s_pos;
    }
    __syncthreads();
    if (tid == 0) {
        float mn = red[0][0], sn = red[0][1], mp = red[0][2], sp = red[0][3];
        #pragma unroll
        for (int w = 1; w < NWAVES; ++w) {
            lse_combine(mn, sn, red[w][0], red[w][1]);
            lse_combine(mp, sp, red[w][2], red[w][3]);
        }
        // s == 0 <=> stream had no elements (matches has_pos/has_neg validity).
        const bool valid = (sn > 0.0f) && (sp > 0.0f);
        float loss = 0.0f;
        if (valid) {
            const float logS = (mn + logf(sn)) + (mp + logf(sp));
            loss = (logS > 30.0f) ? logS : log1pf(__expf(logS));
        }
        row_loss[row]  = loss;
        row_valid[row] = valid ? 1.0f : 0.0f;
    }
}

// Deterministic final reduction: fixed per-thread strided sums + fixed LDS tree.
__global__ void __launch_bounds__(256) lesp_final_kernel(
    const float* __restrict__ row_loss,
    const float* __restrict__ row_valid,
    float* __restrict__ out, int nrows)
{
    __shared__ float sl[256];
    __shared__ float sv[256];
    const int tid = threadIdx.x;
    float tl = 0.0f, tv = 0.0f;
    for (int i = tid; i < nrows; i += 256) {
        tl += row_loss[i];
        tv += row_valid[i];
    }
    sl[tid] = tl; sv[tid] = tv;
    __syncthreads();
    #pragma unroll
    for (int s = 128; s > 0; s >>= 1) {
        if (tid < s) { sl[tid] += sl[tid + s]; sv[tid] += sv[tid + s]; }
        __syncthreads();
    }
    if (tid == 0) out[0] = (sv[0] > 0.0f) ? (sl[0] / sv[0]) : 0.0f;
}

extern "C" void kernel_launch(void* const* d_in, const int* in_sizes, int n_in,
                              void* d_out, int out_size, void* d_ws, size_t ws_size,
                              hipStream_t stream) {
    const float* logits  = (const float*)d_in[0];
    const int*   targets = (const int*)d_in[1];
    const int B = in_sizes[0] / LCOLS;   // 8192 for the reference shapes

    float* row_loss  = (float*)d_ws;
    float* row_valid = row_loss + B;

    lesp_row_kernel<<<B, TPB, 0, stream>>>(logits, targets, row_loss, row_valid);
    lesp_final_kernel<<<1, 256, 0, stream>>>(row_loss, row_valid, (float*)d_out, B);
}